// HierarchicalRouter_9620726743476
// MI455X (gfx1250) — compile-verified
//
#include <hip/hip_runtime.h>

typedef __attribute__((ext_vector_type(2))) float v2f;
typedef __attribute__((ext_vector_type(8))) float v8f;

#define D_DIM        2048
#define WSTRIDE      2052   // 2048 + 4-word pad: conflict-free ds_load_b64 across 16 rows
#define SSTRIDE      20     // 16 + 4 pad: conflict-free float4 reads, 80B keeps 16B align
#define TOK_PER_WAVE 16
#define WAVES        8
#define TOK_PER_BLOCK (TOK_PER_WAVE * WAVES)

__global__ __launch_bounds__(256)
void hier_router_kernel(const float* __restrict__ x,
                        const float* __restrict__ W1,
                        const float* __restrict__ W2,
                        int*   __restrict__ out_idx,
                        float* __restrict__ out_w,
                        int num_tokens)
{
    // 16 rows (8 from W1, 8 from W2) of 2048, padded -> ~128.3 KB
    __shared__ float sW[16 * WSTRIDE];
    // per-wave 16x16 score tiles, padded -> 10 KB
    __shared__ float sSc[WAVES * TOK_PER_WAVE * SSTRIDE];

    const int tid  = threadIdx.x;
    const int lane = tid & 31;
    const int wave = tid >> 5;

    // ---- stage W = concat(W1, W2) into LDS with padded rows ----
    for (int i = tid; i < (16 * D_DIM) / 4; i += 256) {
        const int row  = i >> 9;             // i / (2048/4)
        const int col  = (i & 511) << 2;     // element column
        const float* src = (row < 8) ? (W1 + (row << 11) + col)
                                     : (W2 + ((row - 8) << 11) + col);
        *(float4*)&sW[row * WSTRIDE + col] = *(const float4*)src;
    }
    __syncthreads();

    const int tokBase = blockIdx.x * TOK_PER_BLOCK + wave * TOK_PER_WAVE;
    const int tRow    = lane & 15;           // token row in tile / N index for B
    const int kOff    = (lane >> 4) << 1;    // lanes 0-15: K=0..1, lanes 16-31: K=2..3
    const int last    = num_tokens - 1;
    int tokLd = tokBase + tRow; if (tokLd > last) tokLd = last;  // clamp, keep EXEC full

    const float* aPtr = x  + (size_t)tokLd * D_DIM + kOff;       // global, b64/step
    const float* bPtr = sW + tRow * WSTRIDE + kOff;              // LDS, b64/step

    // ---- GEMM: accumulate 16x16 f32 tile over K=2048 ----
    v8f c = {};
    #pragma unroll 8
    for (int k = 0; k < D_DIM; k += 4) {
        v2f a = *(const v2f*)(aPtr + k);
        v2f b = *(const v2f*)(bPtr + k);
        c = __builtin_amdgcn_wmma_f32_16x16x4_f32(false, a, false, b,
                                                  (short)0, c, false, false);
    }

    // ---- scatter C tile to LDS as [token][16 scores] ----
    {
        const int mBase = (lane >> 4) << 3;  // VGPR j: rows j (lo lanes) and j+8 (hi lanes)
        const int n     = lane & 15;
        float* dst = &sSc[wave * TOK_PER_WAVE * SSTRIDE];
        #pragma unroll
        for (int j = 0; j < 8; ++j)
            dst[(mBase + j) * SSTRIDE + n] = c[j];
    }
    __syncthreads();

    // ---- epilogue: one token per lane (lanes 0-15) ----
    if (lane < 16) {
        const int tok = tokBase + lane;
        const float* sS = &sSc[(wave * TOK_PER_WAVE + lane) * SSTRIDE];
        float4 q0 = *(const float4*)(sS + 0);
        float4 q1 = *(const float4*)(sS + 4);
        float4 q2 = *(const float4*)(sS + 8);
        float4 q3 = *(const float4*)(sS + 12);
        float g[8] = {q0.x, q0.y, q0.z, q0.w, q1.x, q1.y, q1.z, q1.w};
        float e[8] = {q2.x, q2.y, q2.z, q2.w, q3.x, q3.y, q3.z, q3.w};

        // top-2 groups (strict > keeps earliest index on ties, like jax top_k)
        int gi0 = 0; float gv0 = g[0];
        #pragma unroll
        for (int i = 1; i < 8; ++i) if (g[i] > gv0) { gv0 = g[i]; gi0 = i; }
        int gi1 = (gi0 == 0) ? 1 : 0; float gv1 = g[gi1];
        #pragma unroll
        for (int i = 0; i < 8; ++i) if (i != gi0 && g[i] > gv1) { gv1 = g[i]; gi1 = i; }

        // top-2 experts
        int ei0 = 0; float ev0 = e[0];
        #pragma unroll
        for (int i = 1; i < 8; ++i) if (e[i] > ev0) { ev0 = e[i]; ei0 = i; }
        int ei1 = (ei0 == 0) ? 1 : 0; float ev1 = e[ei1];
        #pragma unroll
        for (int i = 0; i < 8; ++i) if (i != ei0 && e[i] > ev1) { ev1 = e[i]; ei1 = i; }

        // combine + softmax over the 4 slots
        const float s0 = gv0 + ev0, s1 = gv0 + ev1, s2 = gv1 + ev0, s3 = gv1 + ev1;
        const float m  = fmaxf(fmaxf(s0, s1), fmaxf(s2, s3));
        const float w0 = __expf(s0 - m), w1 = __expf(s1 - m);
        const float w2 = __expf(s2 - m), w3 = __expf(s3 - m);
        const float inv = 1.0f / (w0 + w1 + w2 + w3);

        if (tok < num_tokens) {
            int4 iv; iv.x = (gi0 << 3) + ei0; iv.y = (gi0 << 3) + ei1;
                     iv.z = (gi1 << 3) + ei0; iv.w = (gi1 << 3) + ei1;
            *(int4*)(out_idx + (size_t)tok * 4) = iv;
            float4 wv; wv.x = w0 * inv; wv.y = w1 * inv; wv.z = w2 * inv; wv.w = w3 * inv;
            *(float4*)(out_w + (size_t)tok * 4) = wv;
        }
    }
}

extern "C" void kernel_launch(void* const* d_in, const int* in_sizes, int n_in,
                              void* d_out, int out_size, void* d_ws, size_t ws_size,
                              hipStream_t stream) {
    const float* x  = (const float*)d_in[0];
    const float* W1 = (const float*)d_in[1];
    const float* W2 = (const float*)d_in[2];
    const int num_tokens = in_sizes[0] / D_DIM;          // 4*8192 = 32768
    int*   out_idx = (int*)d_out;                        // first output: int32 indices
    float* out_w   = (float*)d_out + (size_t)num_tokens * 4; // second output: fp32 weights
    const int blocks = (num_tokens + TOK_PER_BLOCK - 1) / TOK_PER_BLOCK;
    hier_router_kernel<<<blocks, 256, 0, stream>>>(x, W1, W2, out_idx, out_w, num_tokens);
}